// S4DKernel_37134287241903
// MI455X (gfx1250) — compile-verified
//
#include <hip/hip_runtime.h>
#include <hip/hip_bf16.h>
#include <math.h>

// ---------------------------------------------------------------------------
// S4D forward (exact chunked SSM formulation, no FFT)
//   BATCH=8, LENGTH=2048, D_MODEL=1024, D_STATE=64, chunk T=16
//   y[b,l,d] = sum_{k<=l} K[d,l-k] x[b,k,d] + D[d] x[b,l,d]
//   split:  intra-chunk 16-tap Toeplitz conv  (WMMA f32 16x16x4, exact)
//         + carried-state term  y += G @ s    (WMMA f32 16x16x4, batch as N)
// ---------------------------------------------------------------------------

#define BATCH    8
#define LENGTH   2048
#define D_MODEL  1024
#define D_STATE  64
#define TCH      16                     // chunk length (== WMMA tile M/N)
#define CHUNKS   (LENGTH / TCH)         // 128
#define COLS     (BATCH * CHUNKS)       // 1024 independent chunk-columns / channel
#define TILES    (COLS / 16)            // 64 column tiles / channel

typedef __attribute__((ext_vector_type(2))) float v2f;
typedef __attribute__((ext_vector_type(8))) float v8f;

#if defined(__HIP_DEVICE_COMPILE__)
#if !__has_builtin(__builtin_amdgcn_wmma_f32_16x16x4_f32)
#error "gfx1250 device pass: __builtin_amdgcn_wmma_f32_16x16x4_f32 not available"
#endif
#endif

__device__ __forceinline__ v8f wmma_f32_16x16x4(v2f a, v2f b, v8f c) {
#if defined(__HIP_DEVICE_COMPILE__)
  // 8 args: (neg_a, A, neg_b, B, c_mod, C, reuse_a, reuse_b)
  return __builtin_amdgcn_wmma_f32_16x16x4_f32(false, a, false, b, (short)0, c,
                                               false, false);
#else
  (void)a; (void)b;
  return c;  // host pass stub; never executed
#endif
}

// ---------------------------------------------------------------------------
// Kernel A: intra-chunk Toeplitz conv + skip connection (writes y).
// grid = D_MODEL, block = 256 (8 waves); taps computed ONCE per channel,
// each wave then processes 8 column tiles reusing its Toeplitz A-fragment.
// ---------------------------------------------------------------------------
__global__ __launch_bounds__(256) void s4d_intra(
    const float* __restrict__ x, const float* __restrict__ log_dt,
    const float* __restrict__ Bmat, const float* __restrict__ Cmat,
    const float* __restrict__ Dvec, float* __restrict__ y)
{
  const int d = blockIdx.x;
  __shared__ float sP[TCH][D_STATE];   // geometric columns W[n]*Abar[n]^j
  __shared__ float sK[TCH];            // first 16 Vandermonde taps of channel d

  const int tid = threadIdx.x;
  const float dt = expf(log_dt[d]);

  // thread n: iterate its geometric sequence (no transcendentals beyond exp)
  if (tid < D_STATE) {
    const float An  = -0.5f * (float)(tid + 1);
    const float dtA = dt * An;
    const float Abar = expf(dtA);
    float q = Cmat[d * D_STATE + tid] *
              (expm1f(dtA) / An) * Bmat[d * D_STATE + tid];   // W = C * Bbar
    #pragma unroll
    for (int j = 0; j < TCH; ++j) { sP[j][tid] = q; q *= Abar; }
  }
  __syncthreads();
  if (tid < TCH) {                     // deterministic fixed-order reduction
    float acc = 0.f;
    #pragma unroll 8
    for (int n = 0; n < D_STATE; ++n) acc += sP[tid][n];
    sK[tid] = acc;                     // K[d, tid]
  }
  __syncthreads();

  const int lane = tid & 31;
  const int wave = tid >> 5;
  const int row  = lane & 15;          // M row of A-frag, N col of B/D frags
  const int hi   = lane >> 4;

  // A-fragments: lower-triangular Toeplitz M[r][c] = K[r-c]; K-dim split 4x4
  v2f afrag[4];
  #pragma unroll
  for (int kk = 0; kk < 4; ++kk) {
    #pragma unroll
    for (int v = 0; v < 2; ++v) {
      const int col = kk * 4 + hi * 2 + v;
      afrag[kk][v] = (row >= col) ? sK[row - col] : 0.f;
    }
  }

  const float Dd = Dvec[d];

  for (int t8 = 0; t8 < TILES / 8; ++t8) {
    const int tile = t8 * 8 + wave;    // 0..63
    // column cc enumerates (batch, chunk)
    const int cc = tile * 16 + row;
    const int b  = cc >> 7;
    const int ch = cc & 127;
    const int base = (b * LENGTH + ch * TCH) * D_MODEL + d;

    v8f acc = {};
    #pragma unroll
    for (int kk = 0; kk < 4; ++kk) {
      v2f bfrag;
      #pragma unroll
      for (int v = 0; v < 2; ++v) {
        const int j = kk * 4 + hi * 2 + v;        // position within chunk
        bfrag[v] = x[base + j * D_MODEL];
      }
      acc = wmma_f32_16x16x4(afrag[kk], bfrag, acc);
    }

    // D-fragment: vgpr v holds output position r = v + hi*8, column = row
    #pragma unroll
    for (int v = 0; v < 8; ++v) {
      const int r   = v + hi * 8;
      const int idx = base + r * D_MODEL;
      y[idx] = acc[v] + Dd * x[idx];             // fuse skip connection
    }
  }
}

// ---------------------------------------------------------------------------
// Kernel B: carried-state contribution. One wave per channel.
// State kept directly in WMMA B-operand fragment layout:
//   slot(kk,v): n = kk*4 + hi*2 + v ; column = lane&15 = batch (cols 8..15 == 0)
// Per chunk: y += G(16x64) @ s  (16 chained wmma), then Horner state update.
// ---------------------------------------------------------------------------
__global__ __launch_bounds__(32) void s4d_state(
    const float* __restrict__ x, const float* __restrict__ log_dt,
    const float* __restrict__ Bmat, const float* __restrict__ Cmat,
    float* __restrict__ y)
{
  const int d    = blockIdx.x;
  const int lane = threadIdx.x;
  const int row  = lane & 15;          // batch column
  const int hi   = lane >> 4;

  __shared__ float sAbar[D_STATE];
  __shared__ float sBbar[D_STATE];
  __shared__ float sX[TCH * BATCH];    // staged x chunk [j][b]

  const float dt = expf(log_dt[d]);
  for (int t = lane; t < D_STATE; t += 32) {
    const float An  = -0.5f * (float)(t + 1);
    const float dtA = dt * An;
    sAbar[t] = expf(dtA);
    sBbar[t] = (expm1f(dtA) / An) * Bmat[d * D_STATE + t];
  }
  __syncthreads();

  // G A-fragments: G[r][n] = C[n] * Abar[n]^(r+1); K-dim (n) split into 16x4
  v2f g[16];
  #pragma unroll
  for (int kk = 0; kk < 16; ++kk) {
    #pragma unroll
    for (int v = 0; v < 2; ++v) {
      const int n = kk * 4 + hi * 2 + v;
      g[kk][v] = Cmat[d * D_STATE + n] * powf(sAbar[n], (float)(row + 1));
    }
  }

  // per-slot constants + state registers (B-fragment layout)
  float cA[32], cB[32], cL[32];
  v2f s[16];
  #pragma unroll
  for (int kk = 0; kk < 16; ++kk) {
    #pragma unroll
    for (int v = 0; v < 2; ++v) {
      const int n = kk * 4 + hi * 2 + v;
      const float a1 = sAbar[n];
      const float a2 = a1 * a1, a4 = a2 * a2, a8 = a4 * a4;
      cA[kk * 2 + v] = a1;
      cB[kk * 2 + v] = sBbar[n];
      cL[kk * 2 + v] = a8 * a8;                  // Abar^16
      s[kk][v] = 0.f;
    }
  }

  for (int ch = 0; ch < CHUNKS; ++ch) {
    const int l0 = ch * TCH;

    // cooperative stage of x chunk (16 x 8 values) into LDS
    #pragma unroll
    for (int k = 0; k < 4; ++k) {
      const int e = lane + k * 32;               // 0..127
      const int j = e >> 3;
      const int b = e & 7;
      sX[e] = x[(b * LENGTH + l0 + j) * D_MODEL + d];
    }
    __syncthreads();

    // y_state tile = G @ s_prev  (s holds state at chunk start)
    v8f acc = {};
    #pragma unroll
    for (int kk = 0; kk < 16; ++kk)
      acc = wmma_f32_16x16x4(g[kk], s[kk], acc);

    // accumulate into y (kernel A already wrote the intra + skip terms)
    if (row < BATCH) {
      #pragma unroll
      for (int v = 0; v < 8; ++v) {
        const int r   = v + hi * 8;
        const int idx = (row * LENGTH + l0 + r) * D_MODEL + d;
        y[idx] += acc[v];
      }
    }

    // this lane's batch column of the chunk
    float xv[TCH];
    #pragma unroll
    for (int j = 0; j < TCH; ++j)
      xv[j] = (row < BATCH) ? sX[j * BATCH + row] : 0.f;

    // s_new = Abar^16 * s + Bbar * sum_j Abar^(15-j) x[j]   (Horner)
    #pragma unroll
    for (int kk = 0; kk < 16; ++kk) {
      #pragma unroll
      for (int v = 0; v < 2; ++v) {
        float h = 0.f;
        #pragma unroll
        for (int j = 0; j < TCH; ++j)
          h = fmaf(cA[kk * 2 + v], h, xv[j]);
        s[kk][v] = fmaf(cL[kk * 2 + v], s[kk][v], cB[kk * 2 + v] * h);
      }
    }
    __syncthreads();                              // before sX is overwritten
  }
}

// ---------------------------------------------------------------------------
extern "C" void kernel_launch(void* const* d_in, const int* in_sizes, int n_in,
                              void* d_out, int out_size, void* d_ws, size_t ws_size,
                              hipStream_t stream) {
  (void)in_sizes; (void)n_in; (void)out_size; (void)d_ws; (void)ws_size;
  const float* x      = (const float*)d_in[0];   // [8, 2048, 1024]
  const float* log_dt = (const float*)d_in[1];   // [1024]
  const float* Bmat   = (const float*)d_in[2];   // [1024, 64]
  const float* Cmat   = (const float*)d_in[3];   // [1024, 64]
  const float* Dvec   = (const float*)d_in[4];   // [1024]
  float* y            = (float*)d_out;           // [8, 2048, 1024]

  s4d_intra<<<dim3(D_MODEL), 256, 0, stream>>>(x, log_dt, Bmat, Cmat, Dvec, y);
  s4d_state<<<dim3(D_MODEL), 32, 0, stream>>>(x, log_dt, Bmat, Cmat, y);
}